// GedLayer_53352083750916
// MI455X (gfx1250) — compile-verified
//
#include <hip/hip_runtime.h>

// GED forward for n=m=95 (n1=m1=96). Single-workgroup, LDS-resident,
// WMMA f32 16x16x4 for the two 96x96x96 GEMMs per edge-label bucket.

typedef float v2f __attribute__((ext_vector_type(2)));
typedef float v8f __attribute__((ext_vector_type(8)));

#define NL    95          // nodes per graph
#define NR    96          // n1 = m1
#define SSTR  97          // padded LDS row stride (floats) -> conflict-friendly
#define NELEM (NR*NR)     // 9216

__global__ __launch_bounds__(256) void ged_kernel(
    const float* __restrict__ node_weights,   // 46
    const float* __restrict__ edge_weights,   // 4
    const int*   __restrict__ A_g1,           // 95*95
    const int*   __restrict__ A_g2,           // 95*95
    const int*   __restrict__ labels1,        // 95
    const int*   __restrict__ labels2,        // 95
    float* __restrict__ out)                  // 1 (ged)
{
    __shared__ float S[NR * SSTR];            // Sinkhorn matrix
    __shared__ float Q[NR * SSTR];            // temp: M1t(a1) @ S
    __shared__ unsigned char A1b[NR * NR];    // padded adjacency labels, row-major
    __shared__ unsigned char A2b[NR * NR];
    __shared__ float Ftab[16];                // 4x4 edge-cost table
    __shared__ float ncost[100];              // 10x10 node cost table
    __shared__ int   lab1[NL], lab2[NL];
    __shared__ float scl[NR];
    __shared__ float red[256];
    __shared__ float sh_nid;                  // node ins/del cost

    const int tid  = threadIdx.x;
    const int lane = tid & 31;
    const int wave = tid >> 5;
    const int lm   = lane & 15;   // lane row/col within 16
    const int hi   = lane >> 4;   // half-wave select (K pair 0/1 vs 2/3)

    // ---- Phase 0: cost tables (tiny; thread 0) ----
    if (tid == 0) {
        float ce[4];
        for (int i = 0; i < 4; ++i) { float v = edge_weights[i]; ce[i] = v > 0.f ? v : 0.f; }
        const float eid = ce[3];
        // edge_costs 3x3 symmetric, zero diag, triu order (0,1),(0,2),(1,2)
        float ec[3][3] = { {0.f, ce[0], ce[1]}, {ce[0], 0.f, ce[2]}, {ce[1], ce[2], 0.f} };
        Ftab[0] = 0.f;
        for (int a = 1; a < 4; ++a) { Ftab[a] = eid; Ftab[a * 4] = eid; }
        for (int a = 1; a < 4; ++a)
            for (int b = 1; b < 4; ++b) Ftab[a * 4 + b] = ec[a - 1][b - 1];

        float nid = node_weights[45]; nid = nid > 0.f ? nid : 0.f;
        sh_nid = nid;
        for (int i = 0; i < 100; ++i) ncost[i] = 0.f;
        int idx = 0;
        for (int a = 0; a < 10; ++a)
            for (int b = a + 1; b < 10; ++b) {
                float v = node_weights[idx++]; v = v > 0.f ? v : 0.f;
                ncost[a * 10 + b] = v;
                ncost[b * 10 + a] = v;
            }
    }

    // ---- padded adjacency (labels 0..3) + node labels into LDS ----
    for (int p = tid; p < NELEM; p += 256) {
        int r = p / NR, c = p % NR;
        A1b[p] = (r < NL && c < NL) ? (unsigned char)A_g1[r * NL + c] : (unsigned char)0;
        A2b[p] = (r < NL && c < NL) ? (unsigned char)A_g2[r * NL + c] : (unsigned char)0;
    }
    for (int p = tid; p < NL; p += 256) { lab1[p] = labels1[p]; lab2[p] = labels2[p]; }
    __syncthreads();

    auto d2 = [&](int j, int k) -> float {
        if (j < NL && k < NL) return ncost[lab1[j] * 10 + lab2[k]];
        if (j == NL && k == NL) return 0.f;
        return sh_nid;
    };

    // ---- S0 = exp(-0.5 * D2) ----
    for (int p = tid; p < NELEM; p += 256) {
        int j = p / NR, k = p % NR;
        S[j * SSTR + k] = __expf(-0.5f * d2(j, k));
    }
    __syncthreads();

    // ---- Sinkhorn, 10 iterations (row scale then col scale; last idx pinned) ----
    for (int it = 0; it < 10; ++it) {
        if (tid < NR) {
            float s = 0.f;
            for (int k = 0; k < NR; ++k) s += S[tid * SSTR + k];
            scl[tid] = (tid == NR - 1) ? 1.0f : (1.0f / s);
        }
        __syncthreads();
        for (int p = tid; p < NELEM; p += 256) { int j = p / NR, k = p % NR; S[j * SSTR + k] *= scl[j]; }
        __syncthreads();
        if (tid < NR) {
            float s = 0.f;
            for (int j = 0; j < NR; ++j) s += S[j * SSTR + tid];
            scl[tid] = (tid == NR - 1) ? 1.0f : (1.0f / s);
        }
        __syncthreads();
        for (int p = tid; p < NELEM; p += 256) { int j = p / NR, k = p % NR; S[j * SSTR + k] *= scl[k]; }
        __syncthreads();
    }

    // ---- Quadratic term via WMMA f32 16x16x4 ----
    // quad = sum_{a1} sum_{k,l} (S^T (M1t_{a1} S))[k,l] * F[a1][A2[k,l]]
    float quad = 0.f;
    for (int a1 = 0; a1 < 4; ++a1) {
        // GEMM 1: Q = M1t(a1) @ S, M1t[i][j] = (A1[j][i]==a1)
        for (int t = wave; t < 36; t += 8) {
            int tm = t / 6, tn = t % 6;
            v8f acc = {0.f,0.f,0.f,0.f,0.f,0.f,0.f,0.f};
            for (int kk = 0; kk < 24; ++kk) {
                int k0 = kk * 4 + 2 * hi;
                v2f a, b;
                a.x = (A1b[(k0 + 0) * NR + tm * 16 + lm] == a1) ? 1.0f : 0.0f;
                a.y = (A1b[(k0 + 1) * NR + tm * 16 + lm] == a1) ? 1.0f : 0.0f;
                b.x = S[(k0 + 0) * SSTR + tn * 16 + lm];
                b.y = S[(k0 + 1) * SSTR + tn * 16 + lm];
                acc = __builtin_amdgcn_wmma_f32_16x16x4_f32(
                    false, a, false, b, (short)0, acc, false, false);
            }
            for (int r = 0; r < 8; ++r)
                Q[(tm * 16 + r + 8 * hi) * SSTR + tn * 16 + lm] = acc[r];
        }
        __syncthreads();

        // GEMM 2: P = S^T @ Q, folded immediately into the F-weighted reduction
        for (int t = wave; t < 36; t += 8) {
            int tm = t / 6, tn = t % 6;
            v8f acc = {0.f,0.f,0.f,0.f,0.f,0.f,0.f,0.f};
            for (int kk = 0; kk < 24; ++kk) {
                int k0 = kk * 4 + 2 * hi;
                v2f a, b;
                a.x = S[(k0 + 0) * SSTR + tm * 16 + lm];
                a.y = S[(k0 + 1) * SSTR + tm * 16 + lm];
                b.x = Q[(k0 + 0) * SSTR + tn * 16 + lm];
                b.y = Q[(k0 + 1) * SSTR + tn * 16 + lm];
                acc = __builtin_amdgcn_wmma_f32_16x16x4_f32(
                    false, a, false, b, (short)0, acc, false, false);
            }
            for (int r = 0; r < 8; ++r) {
                int k = tm * 16 + r + 8 * hi;
                int l = tn * 16 + lm;
                quad += acc[r] * Ftab[a1 * 4 + (int)A2b[k * NR + l]];
            }
        }
        __syncthreads();   // before next a1 overwrites Q
    }

    // ---- diagonal correction + linear term ----
    float lin = 0.f, dg = 0.f;
    for (int p = tid; p < NELEM; p += 256) {
        int j = p / NR, k = p % NR;
        float s = S[j * SSTR + k];
        lin += d2(j, k) * s;
        dg  += s * s * Ftab[(int)A1b[j * NR + j] * 4 + (int)A2b[k * NR + k]];
    }

    // ---- final reduction: ged = 0.5*(quad - dg) + lin ----
    red[tid] = 0.5f * (quad - dg) + lin;
    __syncthreads();
    if (tid == 0) {
        float s = 0.f;
        for (int i = 0; i < 256; ++i) s += red[i];
        out[0] = s;
    }
}

extern "C" void kernel_launch(void* const* d_in, const int* in_sizes, int n_in,
                              void* d_out, int out_size, void* d_ws, size_t ws_size,
                              hipStream_t stream) {
    (void)in_sizes; (void)n_in; (void)out_size; (void)d_ws; (void)ws_size;
    const float* nw = (const float*)d_in[0];
    const float* ew = (const float*)d_in[1];
    const int*   a1 = (const int*)d_in[2];
    const int*   a2 = (const int*)d_in[3];
    const int*   l1 = (const int*)d_in[4];
    const int*   l2 = (const int*)d_in[5];
    float* out = (float*)d_out;
    hipLaunchKernelGGL(ged_kernel, dim3(1), dim3(256), 0, stream,
                       nw, ew, a1, a2, l1, l2, out);
}